// GroupedMultiQuerySelfAttention_78426102825679
// MI455X (gfx1250) — compile-verified
//
#include <hip/hip_runtime.h>

typedef float v2f __attribute__((ext_vector_type(2)));
typedef float v8f __attribute__((ext_vector_type(8)));

// D = A(16x4 f32) * B(4x16 f32) + C(16x16 f32) on the CDNA5 matrix pipe.
__device__ __forceinline__ v8f wmma_f32(v2f a, v2f b, v8f c) {
  return __builtin_amdgcn_wmma_f32_16x16x4_f32(
      /*neg_a=*/false, a, /*neg_b=*/false, b,
      /*c_mod=*/(short)0, c, /*reuse_a=*/false, /*reuse_b=*/false);
}

// CDNA5 async DMA: global -> LDS, 128b per lane, tracked by ASYNCcnt.
__device__ __forceinline__ void async_b128(unsigned lds_byte_off, const float* g) {
  asm volatile("global_load_async_to_lds_b128 %0, %1, off"
               :: "v"(lds_byte_off), "v"(g)
               : "memory");
}
__device__ __forceinline__ void wait_async0() {
  asm volatile("s_wait_asynccnt 0x0" ::: "memory");
}

// ---------------------------------------------------------------------------
// GEMM: C[M,N] = X[M,K] @ W[K,N] + bias[N]   (fp32, row-major)
// 256 threads = 8 waves; block tile 128x128; wave tile 64x32 (4x2 WMMA accums)
// Double-buffered LDS fed by async B128 copies.
// ---------------------------------------------------------------------------
#define GTM 128
#define GTN 128
#define GTK 16
#define XS_STR (GTK + 4)   // 20 floats: 16B-aligned rows, conflict-free A-frag reads
#define WS_STR (GTN + 4)   // 132 floats

__global__ __launch_bounds__(256) void gemm_bias_wmma(
    const float* __restrict__ X, const float* __restrict__ W,
    const float* __restrict__ bias, float* __restrict__ C,
    int M, int N, int K) {
  __shared__ float Xs[2][GTM][XS_STR];
  __shared__ float Ws[2][GTK][WS_STR];

  const int t    = threadIdx.x;
  const int lane = t & 31;
  const int wave = t >> 5;
  const int lm   = lane & 15;          // col / row-in-tile index
  const int kb   = (lane >> 4) * 2;    // K sub-base for A/B frags
  const int lh8  = (lane >> 4) * 8;    // C/D row offset for upper half-wave

  const int wm = (wave & 1) * 64;      // wave M offset within block tile
  const int wn = (wave >> 1) * 32;     // wave N offset within block tile
  const int bm = blockIdx.y * GTM;
  const int bn = blockIdx.x * GTN;

  auto stage = [&](int sbuf, int k0) {
    // X tile: 128x16 floats = 512 float4, 2 per thread.
#pragma unroll
    for (int i = 0; i < 2; ++i) {
      int q = t + i * 256;
      int r = q >> 2, c4 = q & 3;
      async_b128((unsigned)(size_t)&Xs[sbuf][r][c4 * 4],
                 &X[(size_t)(bm + r) * K + (k0 + c4 * 4)]);
    }
    // W tile: 16x128 floats = 512 float4, 2 per thread.
#pragma unroll
    for (int i = 0; i < 2; ++i) {
      int q = t + i * 256;
      int r = q >> 5, c4 = q & 31;
      async_b128((unsigned)(size_t)&Ws[sbuf][r][c4 * 4],
                 &W[(size_t)(k0 + r) * N + (bn + c4 * 4)]);
    }
  };

  v8f acc[4][2] = {};

  int buf = 0;
  stage(0, 0);
  wait_async0();
  __syncthreads();

  for (int k0 = 0; k0 < K; k0 += GTK) {
    if (k0 + GTK < K) stage(buf ^ 1, k0 + GTK);  // async prefetch of next slice

    // Preload all fragments for this slice, then fire all WMMAs.
    v2f af[4][4], bf[2][4];
#pragma unroll
    for (int kk = 0; kk < 4; ++kk) {
      const int kbase = kk * 4 + kb;
#pragma unroll
      for (int i = 0; i < 4; ++i) {
        af[i][kk].x = Xs[buf][wm + i * 16 + lm][kbase];
        af[i][kk].y = Xs[buf][wm + i * 16 + lm][kbase + 1];
      }
#pragma unroll
      for (int j = 0; j < 2; ++j) {
        bf[j][kk].x = Ws[buf][kbase][wn + j * 16 + lm];
        bf[j][kk].y = Ws[buf][kbase + 1][wn + j * 16 + lm];
      }
    }
#pragma unroll
    for (int kk = 0; kk < 4; ++kk)
#pragma unroll
      for (int i = 0; i < 4; ++i)
#pragma unroll
        for (int j = 0; j < 2; ++j)
          acc[i][j] = wmma_f32(af[i][kk], bf[j][kk], acc[i][j]);

    wait_async0();      // our async writes into buf^1 have landed in LDS
    __syncthreads();    // everyone done reading buf / writing buf^1
    buf ^= 1;
  }

#pragma unroll
  for (int i = 0; i < 4; ++i) {
#pragma unroll
    for (int j = 0; j < 2; ++j) {
      const int col = bn + wn + j * 16 + lm;
      const float bv = bias[col];
#pragma unroll
      for (int g = 0; g < 8; ++g) {
        const int row = bm + wm + i * 16 + lh8 + g;
        C[(size_t)row * N + col] = acc[i][j][g] + bv;
      }
    }
  }
}

// ---------------------------------------------------------------------------
// Fused GQA attention (flash style, fp32 WMMA), double-buffered async K/V.
// grid = (n/64, 16 heads, 2 batches); 128 threads = 4 waves; wave owns 16 rows.
// K kept row-major in LDS: B-frag for Q*K^T is Kr[n][kbase..kbase+1] (contig).
// ---------------------------------------------------------------------------
#define NSEQ   2048
#define DMODEL 1024
#define DKV    256
#define DH     64
#define AT_STR (DH + 4)   // 68 floats

__global__ __launch_bounds__(128) void attn_fused(
    const float* __restrict__ Q, const float* __restrict__ Kp,
    const float* __restrict__ Vp, float* __restrict__ O) {
  __shared__ float Kr[2][DH][AT_STR];     // K tile, row-major: Kr[n][k]
  __shared__ float Vs[2][DH][AT_STR];     // V tile, row-major: Vs[n][k]
  __shared__ float Ps[4][16][AT_STR];     // per-wave P slab for A-frag relayout

  const int t    = threadIdx.x;
  const int lane = t & 31;
  const int wave = t >> 5;
  const int lm   = lane & 15;
  const int kb   = (lane >> 4) * 2;
  const int lh8  = (lane >> 4) * 8;

  const int qh   = blockIdx.y;       // 0..15
  const int grp  = qh >> 2;          // KV group (4 heads per group)
  const int bat  = blockIdx.z;
  const int qrow = blockIdx.x * 64 + wave * 16;

  auto stageKV = [&](int sbuf, int kv0) {
    // 64x64 floats each = 1024 float4 per tensor; 8 per thread per tensor.
#pragma unroll
    for (int i = 0; i < 8; ++i) {
      int q = t + i * 128;
      int n = q >> 4, c4 = q & 15;
      const size_t gb = (size_t)(bat * NSEQ + kv0 + n) * DKV + grp * DH + c4 * 4;
      async_b128((unsigned)(size_t)&Kr[sbuf][n][c4 * 4], &Kp[gb]);
      async_b128((unsigned)(size_t)&Vs[sbuf][n][c4 * 4], &Vp[gb]);
    }
  };

  // Q fragments for this wave's 16 rows, pre-scaled by 1/sqrt(64).
  const float* qptr = Q + (size_t)(bat * NSEQ + qrow + lm) * DMODEL + qh * DH;
  v2f aq[16];
#pragma unroll
  for (int c = 0; c < 16; ++c) {
    aq[c].x = qptr[c * 4 + kb] * 0.125f;
    aq[c].y = qptr[c * 4 + kb + 1] * 0.125f;
  }

  float mrow[8], srow[8];
#pragma unroll
  for (int g = 0; g < 8; ++g) { mrow[g] = -1.0e30f; srow[g] = 0.0f; }
  v8f o[4] = {};

  int buf = 0;
  stageKV(0, 0);
  wait_async0();
  __syncthreads();

  for (int kv0 = 0; kv0 < NSEQ; kv0 += 64) {
    if (kv0 + 64 < NSEQ) stageKV(buf ^ 1, kv0 + 64);

    // S(16x64) = Qscaled @ K^T  (B-frag = contiguous pair from row-major K)
    v8f st[4] = {};
#pragma unroll
    for (int c = 0; c < 16; ++c) {
      const int kbase = c * 4 + kb;
#pragma unroll
      for (int j = 0; j < 4; ++j) {
        v2f bk;
        bk.x = Kr[buf][j * 16 + lm][kbase];
        bk.y = Kr[buf][j * 16 + lm][kbase + 1];
        st[j] = wmma_f32(aq[c], bk, st[j]);
      }
    }

    // Online softmax: rows live across 16-lane halves -> shfl_xor(1,2,4,8).
    float fac[8];
#pragma unroll
    for (int g = 0; g < 8; ++g) {
      float v = fmaxf(fmaxf(st[0][g], st[1][g]), fmaxf(st[2][g], st[3][g]));
#pragma unroll
      for (int off = 1; off < 16; off <<= 1) v = fmaxf(v, __shfl_xor(v, off, 32));
      const float mn = fmaxf(mrow[g], v);
      const float f  = __expf(mrow[g] - mn);
      float rs = 0.0f;
#pragma unroll
      for (int j = 0; j < 4; ++j) {
        const float p = __expf(st[j][g] - mn);
        st[j][g] = p;
        rs += p;
      }
#pragma unroll
      for (int off = 1; off < 16; off <<= 1) rs += __shfl_xor(rs, off, 32);
      mrow[g] = mn;
      srow[g] = srow[g] * f + rs;
      fac[g]  = f;
    }
#pragma unroll
    for (int j = 0; j < 4; ++j)
#pragma unroll
      for (int g = 0; g < 8; ++g) o[j][g] *= fac[g];

    // Re-layout P: C-layout -> per-wave LDS slab -> A-fragments.
#pragma unroll
    for (int j = 0; j < 4; ++j)
#pragma unroll
      for (int g = 0; g < 8; ++g)
        Ps[wave][lh8 + g][j * 16 + lm] = st[j][g];

    // O(16x64) += P(16x64) @ V(64x64)
#pragma unroll
    for (int c = 0; c < 16; ++c) {
      const int kbase = c * 4 + kb;
      v2f ap;
      ap.x = Ps[wave][lm][kbase];
      ap.y = Ps[wave][lm][kbase + 1];
#pragma unroll
      for (int j = 0; j < 4; ++j) {
        v2f bv;
        bv.x = Vs[buf][kbase][j * 16 + lm];
        bv.y = Vs[buf][kbase + 1][j * 16 + lm];
        o[j] = wmma_f32(ap, bv, o[j]);
      }
    }

    wait_async0();
    __syncthreads();
    buf ^= 1;
  }

  // Normalize and store (same [b, n, G, hpg, dh] packing as q -> flat [.,1024]).
#pragma unroll
  for (int j = 0; j < 4; ++j) {
#pragma unroll
    for (int g = 0; g < 8; ++g) {
      const int row = qrow + lh8 + g;
      const int col = qh * DH + j * 16 + lm;
      O[(size_t)(bat * NSEQ + row) * DMODEL + col] = o[j][g] / srow[g];
    }
  }
}

// ---------------------------------------------------------------------------
extern "C" void kernel_launch(void* const* d_in, const int* in_sizes, int n_in,
                              void* d_out, int out_size, void* d_ws, size_t ws_size,
                              hipStream_t stream) {
  (void)in_sizes; (void)n_in; (void)out_size; (void)ws_size;
  const float* x  = (const float*)d_in[0];
  const float* Wq = (const float*)d_in[1];
  const float* bq = (const float*)d_in[2];
  const float* Wk = (const float*)d_in[3];
  const float* bk = (const float*)d_in[4];
  const float* Wv = (const float*)d_in[5];
  const float* bv = (const float*)d_in[6];
  const float* Wo = (const float*)d_in[7];
  const float* bo = (const float*)d_in[8];
  float* out = (float*)d_out;

  const int BN = 2 * NSEQ;  // 4096 rows
  float* Qp = (float*)d_ws;                       // 4096 x 1024
  float* Kq = Qp + (size_t)BN * DMODEL;           // 4096 x 256
  float* Vq = Kq + (size_t)BN * DKV;              // 4096 x 256
  float* Ap = Vq + (size_t)BN * DKV;              // 4096 x 1024

  const dim3 blk(256);
  gemm_bias_wmma<<<dim3(DMODEL / GTN, BN / GTM), blk, 0, stream>>>(
      x, Wq, bq, Qp, BN, DMODEL, DMODEL);
  gemm_bias_wmma<<<dim3(DKV / GTN, BN / GTM), blk, 0, stream>>>(
      x, Wk, bk, Kq, BN, DKV, DMODEL);
  gemm_bias_wmma<<<dim3(DKV / GTN, BN / GTM), blk, 0, stream>>>(
      x, Wv, bv, Vq, BN, DKV, DMODEL);

  attn_fused<<<dim3(NSEQ / 64, 16, 2), dim3(128), 0, stream>>>(Qp, Kq, Vq, Ap);

  gemm_bias_wmma<<<dim3(DMODEL / GTN, BN / GTM), blk, 0, stream>>>(
      Ap, Wo, bo, out, BN, DMODEL, DMODEL);
}